// GCNRecommender_70480413327754
// MI455X (gfx1250) — compile-verified
//
#include <hip/hip_runtime.h>

typedef float v2f __attribute__((ext_vector_type(2)));
typedef float v8f __attribute__((ext_vector_type(8)));

#define DIM 64

// ---------------- degree / dinv ----------------
__global__ void k_fill1(float* __restrict__ deg, int n) {
    int i = blockIdx.x * blockDim.x + threadIdx.x;
    if (i < n) deg[i] = 1.0f;   // self-loop contributes 1
}

__global__ void k_count(const int* __restrict__ col, float* __restrict__ deg, long long E) {
    long long e = (long long)blockIdx.x * blockDim.x + threadIdx.x;
    if (e < E) atomicAdd(&deg[col[e]], 1.0f);
}

__global__ void k_rsqrt(float* __restrict__ deg, int n) {
    int i = blockIdx.x * blockDim.x + threadIdx.x;
    if (i < n) deg[i] = rsqrtf(deg[i]);   // deg >= 1 always
}

// ---------------- concat embeddings ----------------
__global__ void k_concat(const float* __restrict__ ue, const float* __restrict__ je,
                         float* __restrict__ X, long long uElems, long long total) {
    long long t = (long long)blockIdx.x * blockDim.x + threadIdx.x;
    if (t < total) X[t] = (t < uElems) ? ue[t] : je[t - uElems];
}

// ---------------- H = X @ W via V_WMMA_F32_16X16X4_F32 ----------------
// One wave per 16x16 output tile; K=64 in 16 steps of 4.
// A (16x4 f32):  lane= M (lane&15); VGPR v holds K = 2*(lane>>4) + v
// B (4x16 f32):  lane= N (lane&15); VGPR v holds K = 2*(lane>>4) + v
// C/D (16x16):   lane= N (lane&15); VGPR v holds M = v + 8*(lane>>4)
__global__ void __launch_bounds__(256)
k_gemm64_wmma(const float* __restrict__ X, const float* __restrict__ W,
              float* __restrict__ H, int nrows) {
    int wave = threadIdx.x >> 5;
    int lane = threadIdx.x & 31;
    long long tile = (long long)blockIdx.x * 8 + wave;   // (row_tile, col_tile) pairs
    long long m0   = (tile >> 2) << 4;                   // row tile * 16
    int       n0   = (int)(tile & 3) << 4;               // col tile * 16
    if (m0 >= (long long)nrows) return;                  // wave-uniform: EXEC stays all-ones

    int mr = lane & 15;
    int hi = lane >> 4;                                  // half-select: K offset 0 or 2

    const float* xrow = X + (m0 + mr) * DIM;             // A: row per lane
    const float* wcol = W + n0 + mr;                     // B: column per lane
    v8f acc = {};
#pragma unroll
    for (int k0 = 0; k0 < DIM; k0 += 4) {
        int ka = k0 + 2 * hi;
        v2f a, b;
        a.x = xrow[ka];
        a.y = xrow[ka + 1];
        b.x = wcol[(long long)ka * DIM];
        b.y = wcol[(long long)(ka + 1) * DIM];
        acc = __builtin_amdgcn_wmma_f32_16x16x4_f32(
            /*neg_a=*/false, a, /*neg_b=*/false, b,
            /*c_mod=*/(short)0, acc, /*reuse_a=*/false, /*reuse_b=*/false);
    }
    float* o = H + (m0 + 8 * hi) * DIM + n0 + mr;
#pragma unroll
    for (int v = 0; v < 8; ++v) o[(long long)v * DIM] = acc[v];
}

// ---------------- edge scatter: agg[col] += H[row] * dinv[row]*dinv[col] ----------------
__global__ void k_scatter(const float* __restrict__ H, const int* __restrict__ rowIdx,
                          const int* __restrict__ colIdx, const float* __restrict__ dinv,
                          float* __restrict__ agg, long long E) {
    long long t = (long long)blockIdx.x * blockDim.x + threadIdx.x;
    long long e = t >> 6;
    int d = (int)(t & 63);
    if (e >= E) return;
    int r = rowIdx[e];
    int c = colIdx[e];
    float nrm = dinv[r] * dinv[c];
    atomicAdd(&agg[(long long)c * DIM + d], H[(long long)r * DIM + d] * nrm);
}

// ---------------- finalize: out = [relu](agg + H*dinv^2 + b) ----------------
__global__ void k_finalize(const float* __restrict__ agg, const float* __restrict__ H,
                           const float* __restrict__ dinv, const float* __restrict__ bias,
                           float* __restrict__ out, long long total, int doRelu) {
    long long t = (long long)blockIdx.x * blockDim.x + threadIdx.x;
    if (t >= total) return;
    long long i = t >> 6;
    int d = (int)(t & 63);
    float di = dinv[i];
    float v = agg[t] + H[t] * di * di + bias[d];
    out[t] = (doRelu && v < 0.0f) ? 0.0f : v;
}

// ---------------- predict: one wave per (user, job) pair ----------------
__global__ void k_predict(const float* __restrict__ X2, const int* __restrict__ uidx,
                          const int* __restrict__ jidx, const float* __restrict__ pw,
                          const float* __restrict__ pb, float* __restrict__ out,
                          int B, int numUsers) {
    int g = (blockIdx.x * blockDim.x + threadIdx.x) >> 5;
    int lane = threadIdx.x & 31;
    if (g >= B) return;
    long long u = (long long)uidx[g];
    long long j = (long long)numUsers + jidx[g];
    float s = X2[u * DIM + lane]       * pw[lane]
            + X2[u * DIM + lane + 32]  * pw[lane + 32]
            + X2[j * DIM + lane]       * pw[64 + lane]
            + X2[j * DIM + lane + 32]  * pw[96 + lane];
#pragma unroll
    for (int off = 16; off > 0; off >>= 1) s += __shfl_xor(s, off, 32);
    if (lane == 0) out[g] = s + pb[0];
}

extern "C" void kernel_launch(void* const* d_in, const int* in_sizes, int n_in,
                              void* d_out, int out_size, void* d_ws, size_t ws_size,
                              hipStream_t stream) {
    const int*   edge = (const int*)d_in[0];     // (2, E)
    const int*   uidx = (const int*)d_in[1];     // (B,)
    const int*   jidx = (const int*)d_in[2];     // (B,)
    const float* ue   = (const float*)d_in[3];   // (U, 64)
    const float* je   = (const float*)d_in[4];   // (J, 64)
    const float* W1   = (const float*)d_in[5];
    const float* b1   = (const float*)d_in[6];
    const float* W2   = (const float*)d_in[7];
    const float* b2   = (const float*)d_in[8];
    const float* pw   = (const float*)d_in[9];   // (128, 1)
    const float* pb   = (const float*)d_in[10];  // (1,)
    float* out = (float*)d_out;

    long long Ecnt = (long long)in_sizes[0] / 2;
    int Bcnt = in_sizes[1];
    int U = in_sizes[3] / DIM;
    int J = in_sizes[4] / DIM;
    int N = U + J;
    const int* rowIdx = edge;
    const int* colIdx = edge + Ecnt;

    // workspace carve-up
    char* ws = (char*)d_ws;
    float* dinv = (float*)ws;
    size_t off = (((size_t)N * sizeof(float)) + 255) & ~(size_t)255;
    float* X   = (float*)(ws + off); off += (size_t)N * DIM * sizeof(float);
    float* H   = (float*)(ws + off); off += (size_t)N * DIM * sizeof(float);
    float* AGG = (float*)(ws + off);

    long long nd  = (long long)N * DIM;       // 12.8M
    long long sth = Ecnt * DIM;               // 204.8M scatter threads
    int tiles = (N / 16) * 4;                 // 16x16 output tiles per GEMM

    // degrees -> dinv
    k_fill1 <<<(N + 255) / 256, 256, 0, stream>>>(dinv, N);
    k_count <<<(unsigned)((Ecnt + 255) / 256), 256, 0, stream>>>(colIdx, dinv, Ecnt);
    k_rsqrt <<<(N + 255) / 256, 256, 0, stream>>>(dinv, N);

    // X = concat(user_emb, job_emb)
    k_concat<<<(unsigned)((nd + 255) / 256), 256, 0, stream>>>(ue, je, X, (long long)U * DIM, nd);

    // ---- layer 1: relu(D^-1/2 (A+I) D^-1/2 (X W1) + b1) ----
    hipMemsetAsync(AGG, 0, (size_t)nd * sizeof(float), stream);
    k_gemm64_wmma<<<(tiles + 7) / 8, 256, 0, stream>>>(X, W1, H, N);
    k_scatter<<<(unsigned)((sth + 255) / 256), 256, 0, stream>>>(H, rowIdx, colIdx, dinv, AGG, Ecnt);
    k_finalize<<<(unsigned)((nd + 255) / 256), 256, 0, stream>>>(AGG, H, dinv, b1, X, nd, 1);

    // ---- layer 2: same, no relu ----
    hipMemsetAsync(AGG, 0, (size_t)nd * sizeof(float), stream);
    k_gemm64_wmma<<<(tiles + 7) / 8, 256, 0, stream>>>(X, W2, H, N);
    k_scatter<<<(unsigned)((sth + 255) / 256), 256, 0, stream>>>(H, rowIdx, colIdx, dinv, AGG, Ecnt);
    k_finalize<<<(unsigned)((nd + 255) / 256), 256, 0, stream>>>(AGG, H, dinv, b2, X, nd, 0);

    // ---- prediction head ----
    k_predict<<<(Bcnt * 32 + 255) / 256, 256, 0, stream>>>(X, uidx, jidx, pw, pb, out, Bcnt, U);
}